// DiffIouRotated_81862076662285
// MI455X (gfx1250) — compile-verified
//
#include <hip/hip_runtime.h>
#include <math.h>

#define TPB 256
#define C_EPS 1e-6f

__global__ __launch_bounds__(TPB)
void diff_iou_rotated_kernel(const float* __restrict__ box1,
                             const float* __restrict__ box2,
                             float* __restrict__ out,
                             int total)
{
    __shared__ float sA[TPB * 5];
    __shared__ float sB[TPB * 5];

    const int tid = threadIdx.x;
    const int block_start = blockIdx.x * TPB;
    const int p = block_start + tid;

    // ---- Stage this block's 2 x 1280 floats into LDS via gfx1250 async copies
    // (GLOBAL_LOAD_ASYNC_TO_LDS_B32, GVS mode: saddr base + vgpr byte offset).
    {
        const float* gA = box1 + (long long)block_start * 5;
        const float* gB = box2 + (long long)block_start * 5;
        const int limit = total * 5 - block_start * 5;   // valid floats in slice
        unsigned ldsA = (unsigned)(size_t)(&sA[0]);      // low 32 bits of flat = LDS offset
        unsigned ldsB = (unsigned)(size_t)(&sB[0]);
#pragma unroll
        for (int k = 0; k < 5; ++k) {
            int e = k * TPB + tid;                       // float index within slice
            if (e < limit) {
                unsigned voff = (unsigned)e * 4u;
                unsigned la = ldsA + voff;
                unsigned lb = ldsB + voff;
                asm volatile("global_load_async_to_lds_b32 %0, %1, %2"
                             :: "v"(la), "v"(voff), "s"(gA) : "memory");
                asm volatile("global_load_async_to_lds_b32 %0, %1, %2"
                             :: "v"(lb), "v"(voff), "s"(gB) : "memory");
            }
        }
        asm volatile("s_wait_asynccnt 0" ::: "memory");
    }
    __syncthreads();

    if (p >= total) return;

    float b1[5], b2[5];
#pragma unroll
    for (int j = 0; j < 5; ++j) {
        b1[j] = sA[tid * 5 + j];
        b2[j] = sB[tid * 5 + j];
    }

    // ---- box2corners: rotated = [lx*cos - ly*sin, lx*sin + ly*cos] + center
    const float SX[4] = { 0.5f, -0.5f, -0.5f,  0.5f };
    const float SY[4] = { 0.5f,  0.5f, -0.5f, -0.5f };
    float cx1[4], cy1[4], cx2[4], cy2[4];
    {
        float sn = sinf(b1[4]), cs = cosf(b1[4]);
#pragma unroll
        for (int v = 0; v < 4; ++v) {
            float lx = SX[v] * b1[2], ly = SY[v] * b1[3];
            cx1[v] = lx * cs - ly * sn + b1[0];
            cy1[v] = lx * sn + ly * cs + b1[1];
        }
    }
    {
        float sn = sinf(b2[4]), cs = cosf(b2[4]);
#pragma unroll
        for (int v = 0; v < 4; ++v) {
            float lx = SX[v] * b2[2], ly = SY[v] * b2[3];
            cx2[v] = lx * cs - ly * sn + b2[0];
            cy2[v] = lx * sn + ly * cs + b2[1];
        }
    }

    // ---- candidate vertices: 0..3 = c1, 4..7 = c2, 8..23 = edge intersections
    float vx[24], vy[24];
    unsigned mbits = 0;
#pragma unroll
    for (int v = 0; v < 4; ++v) {
        vx[v]     = cx1[v]; vy[v]     = cy1[v];
        vx[4 + v] = cx2[v]; vy[4 + v] = cy2[v];
    }

    // box_intersection (division-by-zero semantics preserved: masks from den/num,
    // point from den/(num+EPS), masked points zeroed)
#pragma unroll
    for (int i = 0; i < 4; ++i) {
        float X1 = cx1[i],           Y1 = cy1[i];
        float X2 = cx1[(i + 1) & 3], Y2 = cy1[(i + 1) & 3];
#pragma unroll
        for (int j = 0; j < 4; ++j) {
            float X3 = cx2[j],           Y3 = cy2[j];
            float X4 = cx2[(j + 1) & 3], Y4 = cy2[(j + 1) & 3];
            float num  = (X1 - X2) * (Y3 - Y4) - (Y1 - Y2) * (X3 - X4);
            float dent = (X1 - X3) * (Y3 - Y4) - (Y1 - Y3) * (X3 - X4);
            float t0 = dent / num;
            bool mt = (t0 >= 0.0f) && (t0 <= 1.0f);
            float denu = (X1 - X2) * (Y1 - Y3) - (Y1 - Y2) * (X1 - X3);
            float uu = -denu / num;
            bool mu = (uu >= 0.0f) && (uu <= 1.0f);
            bool m = mt && mu;
            float t = dent / (num + C_EPS);
            float px = X1 + t * (X2 - X1);
            float py = Y1 + t * (Y2 - Y1);
            int idx = 8 + i * 4 + j;
            vx[idx] = m ? px : 0.0f;
            vy[idx] = m ? py : 0.0f;
            mbits |= (m ? 1u : 0u) << idx;
        }
    }

    // ---- box1_in_box2 ray-crossing tests (both directions)
#pragma unroll
    for (int v = 0; v < 4; ++v) {
        int cnt = 0;
#pragma unroll
        for (int u = 0; u < 4; ++u) {          // c1[v] vs polygon c2
            float X  = cx2[u] - cx1[v],           Y  = cy2[u] - cy1[v];
            float LX = cx2[(u + 1) & 3] - cx1[v], LY = cy2[(u + 1) & 3] - cy1[v];
            bool m1 = (LY > 0.0f) != (Y > 0.0f);
            float r = X - Y * (LX - X) / (LY - Y);
            cnt += (m1 && (r > 0.0f)) ? 1 : 0;
        }
        if (cnt & 1) mbits |= 1u << v;

        cnt = 0;
#pragma unroll
        for (int u = 0; u < 4; ++u) {          // c2[v] vs polygon c1
            float X  = cx1[u] - cx2[v],           Y  = cy1[u] - cy2[v];
            float LX = cx1[(u + 1) & 3] - cx2[v], LY = cy1[(u + 1) & 3] - cy2[v];
            bool m1 = (LY > 0.0f) != (Y > 0.0f);
            float r = X - Y * (LX - X) / (LY - Y);
            cnt += (m1 && (r > 0.0f)) ? 1 : 0;
        }
        if (cnt & 1) mbits |= 1u << (4 + v);
    }

    // ---- centroid of masked vertices (mean = sum / max(nv,1))
    int nv = __popc(mbits);
    float sxm = 0.0f, sym = 0.0f;
#pragma unroll
    for (int k = 0; k < 24; ++k) {
        float msk = ((mbits >> k) & 1u) ? 1.0f : 0.0f;
        sxm += vx[k] * msk;
        sym += vy[k] * msk;
    }
    float nvf = (float)(nv > 0 ? nv : 1);
    float mx = sxm / nvf;
    float my = sym / nvf;

    // ---- angles (masked -> +inf, matching jnp.where(mask, ang, inf))
    float ang[24];
#pragma unroll
    for (int k = 0; k < 24; ++k) {
        float a = atan2f(vy[k] - my, vx[k] - mx);
        ang[k] = ((mbits >> k) & 1u) ? a : __builtin_inff();
    }

    // ---- stable 9-way selection (strict < with first-index-wins ties ==
    // stable argsort prefix). Track best coords alongside so all indexing is static.
    float selx[9], sely[9];
    unsigned used = 0;
#pragma unroll
    for (int k = 0; k < 9; ++k) {
        float bv = 0.0f, bx = 0.0f, by = 0.0f;
        int bi = -1;
#pragma unroll
        for (int j = 0; j < 24; ++j) {
            bool avail = ((used >> j) & 1u) == 0u;
            if (avail && (bi < 0 || ang[j] < bv)) {
                bv = ang[j]; bi = j; bx = vx[j]; by = vy[j];
            }
        }
        used |= 1u << bi;
        selx[k] = bx;
        sely[k] = by;
    }
    // positions >= nv get the first selected vertex (closes the polygon)
#pragma unroll
    for (int k = 1; k < 9; ++k) {
        if (k >= nv) { selx[k] = selx[0]; sely[k] = sely[0]; }
    }

    // ---- shoelace over 9 points (8 terms), |sum|/2
    float tot = 0.0f;
#pragma unroll
    for (int k = 0; k < 8; ++k) {
        tot += selx[k] * sely[k + 1] - sely[k] * selx[k + 1];
    }
    float inter = fabsf(tot) * 0.5f;

    float area1 = b1[2] * b1[3];
    float area2 = b2[2] * b2[3];
    out[p] = inter / (area1 + area2 - inter);
}

extern "C" void kernel_launch(void* const* d_in, const int* in_sizes, int n_in,
                              void* d_out, int out_size, void* d_ws, size_t ws_size,
                              hipStream_t stream) {
    (void)n_in; (void)d_ws; (void)ws_size; (void)out_size;
    const float* box1 = (const float*)d_in[0];
    const float* box2 = (const float*)d_in[1];
    float* out = (float*)d_out;
    int total = in_sizes[0] / 5;                 // B*N box pairs
    int blocks = (total + TPB - 1) / TPB;
    diff_iou_rotated_kernel<<<blocks, TPB, 0, stream>>>(box1, box2, out, total);
}